// SAModule_33397665694062
// MI455X (gfx1250) — compile-verified
//
#include <hip/hip_runtime.h>
#include <hip/hip_bf16.h>
#include <math.h>

// ---------------- problem constants (match reference) ----------------
#define BB 8
#define PP 4096
#define FF 64
#define SS 1024            // P * 0.25
#define KK 128
#define R2 0.04f           // R = 0.2
#define BS (BB * SS)       // 8192 centers total
#define LDH 68             // 67 features padded to 68 (multiple of 4 for WMMA K)
#define C0 128
#define C1 256
#define C2 256

typedef __attribute__((ext_vector_type(2))) float v2f;
typedef __attribute__((ext_vector_type(8))) float v8f;

// ---------------- 1) Farthest point sampling: one block per cloud ----------------
__global__ void fps_kernel(const float* __restrict__ pos, int* __restrict__ idx) {
    const int b    = blockIdx.x;
    const int tid  = threadIdx.x;
    const int lane = tid & 31;
    const int wave = tid >> 5;
    const int T    = blockDim.x;          // 1024
    const int NW   = T >> 5;              // 32 waves

    __shared__ float dist[PP];            // 16 KB
    __shared__ float wVal[32];
    __shared__ int   wIdx[32];
    __shared__ float cur[3];
    __shared__ int   curIdx;

    const float* pb = pos + (size_t)b * PP * 3;
    const float x0 = pb[0], y0 = pb[1], z0 = pb[2];

    for (int p = tid; p < PP; p += T) {
        float dx = pb[p * 3 + 0] - x0;
        float dy = pb[p * 3 + 1] - y0;
        float dz = pb[p * 3 + 2] - z0;
        dist[p] = dx * dx + dy * dy + dz * dz;
    }
    if (tid == 0) idx[b * SS] = 0;
    __syncthreads();

    for (int s = 1; s < SS; ++s) {
        // ---- block-wide argmax over dist (first occurrence on ties) ----
        float bv = -1.0f; int bi = 0;
        for (int p = tid; p < PP; p += T) {
            float v = dist[p];
            if (v > bv) { bv = v; bi = p; }
        }
        for (int off = 16; off > 0; off >>= 1) {
            float ov = __shfl_down(bv, off);
            int   oi = __shfl_down(bi, off);
            if (ov > bv || (ov == bv && oi < bi)) { bv = ov; bi = oi; }
        }
        if (lane == 0) { wVal[wave] = bv; wIdx[wave] = bi; }
        __syncthreads();
        if (tid == 0) {
            float gv = wVal[0]; int gi = wIdx[0];
            for (int w = 1; w < NW; ++w) {
                float ov = wVal[w]; int oi = wIdx[w];
                if (ov > gv || (ov == gv && oi < gi)) { gv = ov; gi = oi; }
            }
            idx[b * SS + s] = gi;
            curIdx = gi;
            cur[0] = pb[gi * 3 + 0];
            cur[1] = pb[gi * 3 + 1];
            cur[2] = pb[gi * 3 + 2];
        }
        __syncthreads();
        const float cx = cur[0], cy = cur[1], cz = cur[2];
        for (int p = tid; p < PP; p += T) {
            float dx = pb[p * 3 + 0] - cx;
            float dy = pb[p * 3 + 1] - cy;
            float dz = pb[p * 3 + 2] - cz;
            float nd = dx * dx + dy * dy + dz * dz;
            if (nd < dist[p]) dist[p] = nd;
        }
        __syncthreads();
        (void)curIdx;
    }
}

// ---------------- 2) gather centers, emit new_pos / new_batch ----------------
__global__ void centers_kernel(const float* __restrict__ pos, const int* __restrict__ idx,
                               float* __restrict__ centers,
                               float* __restrict__ out_pos, float* __restrict__ out_batch) {
    int t = blockIdx.x * blockDim.x + threadIdx.x;
    if (t >= BS) return;
    int b = t / SS;
    int i = idx[t];
    const float* p = pos + ((size_t)b * PP + i) * 3;
    float x = p[0], y = p[1], z = p[2];
    centers[t * 3 + 0] = x; centers[t * 3 + 1] = y; centers[t * 3 + 2] = z;
    out_pos[t * 3 + 0] = x; out_pos[t * 3 + 1] = y; out_pos[t * 3 + 2] = z;
    out_batch[t] = (float)b;
}

// ---------------- 3) fused ball query (first-K in index order) + max pool ----------------
// one wave32 per center; lane covers features {lane, lane+32} plus pos-diff for lane<3
__global__ void ball_max_kernel(const float* __restrict__ x, const float* __restrict__ pos,
                                const float* __restrict__ centers, float* __restrict__ h0) {
    const int lane = threadIdx.x & 31;
    const int c = blockIdx.x * (blockDim.x >> 5) + (threadIdx.x >> 5);
    const int b = c / SS;
    const float cx = centers[c * 3 + 0];
    const float cy = centers[c * 3 + 1];
    const float cz = centers[c * 3 + 2];
    const float cl = (lane < 3) ? centers[c * 3 + lane] : 0.0f;
    const float* pb = pos + (size_t)b * PP * 3;
    const float* xb = x   + (size_t)b * PP * FF;

    float m0 = -INFINITY, m1 = -INFINITY, mp = -INFINITY;
    int count = 0;
    for (int base = 0; base < PP && count < KK; base += 32) {
        int p = base + lane;
        float dx = pb[p * 3 + 0] - cx;
        float dy = pb[p * 3 + 1] - cy;
        float dz = pb[p * 3 + 2] - cz;
        bool inR = (dx * dx + dy * dy + dz * dz) <= R2;
        unsigned mask = (unsigned)__ballot(inR);
        int take = min(__popc(mask), KK - count);
        unsigned mm = mask;
        for (int t = 0; t < take; ++t) {
            int bit = __ffs(mm) - 1; mm &= (mm - 1);
            int nb = base + bit;
            const float* rx = xb + (size_t)nb * FF;
            m0 = fmaxf(m0, rx[lane]);
            m1 = fmaxf(m1, rx[lane + 32]);
            if (lane < 3) mp = fmaxf(mp, pb[nb * 3 + lane] - cl);
        }
        count += take;
    }
    float* hr = h0 + (size_t)c * LDH;
    hr[lane]      = m0;
    hr[32 + lane] = m1;
    if (lane < 3)  hr[64 + lane] = mp;
    if (lane == 3) hr[67] = 0.0f;
}

// ---------------- 4) zero-pad W0: [67,128] -> [68,128] ----------------
__global__ void pad_w0_kernel(const float* __restrict__ W0, float* __restrict__ W0p) {
    int t = blockIdx.x * blockDim.x + threadIdx.x;
    if (t >= LDH * C0) return;
    int row = t / C0, col = t % C0;
    W0p[t] = (row < 67) ? W0[row * C0 + col] : 0.0f;
}

// ---------------- 5) GEMM + bias + ReLU via V_WMMA_F32_16X16X4_F32 ----------------
// one wave32 per 16x16 output tile; K stepped by 4 per the CDNA5 fp32 WMMA layout
__global__ void gemm_bias_relu_wmma(const float* __restrict__ A, int lda,
                                    const float* __restrict__ Bw,
                                    const float* __restrict__ bias,
                                    float* __restrict__ C,
                                    int N, int Kd) {
    const int lane = threadIdx.x & 31;
    const int wid  = blockIdx.x * (blockDim.x >> 5) + (threadIdx.x >> 5);
    const int ntiles = N >> 4;
    const int tm = wid / ntiles;
    const int tn = wid % ntiles;
    const int row = lane & 15;   // A: M index / B,C,D: N index within tile
    const int hi  = lane >> 4;   // K half (A/B), row half offset (C/D)

    const float* Arow = A + (size_t)(tm * 16 + row) * lda;
    const float* Bcol = Bw + (tn * 16 + row);

    v8f acc = {};
    for (int k = 0; k < Kd; k += 4) {
        v2f a, bv;
        a.x  = Arow[k + 2 * hi];
        a.y  = Arow[k + 2 * hi + 1];
        bv.x = Bcol[(size_t)(k + 2 * hi) * N];
        bv.y = Bcol[(size_t)(k + 2 * hi + 1) * N];
        acc = __builtin_amdgcn_wmma_f32_16x16x4_f32(
            /*neg_a=*/false, a, /*neg_b=*/false, bv,
            /*c_mod=*/(short)0, acc, /*reuse_a=*/false, /*reuse_b=*/false);
    }

    const int n = tn * 16 + row;
    const float bz = bias[n];
    float* Crow = C + (size_t)(tm * 16 + hi * 8) * N + n;   // lanes 16-31 cover rows +8
#pragma unroll
    for (int r = 0; r < 8; ++r) {
        Crow[(size_t)r * N] = fmaxf(acc[r] + bz, 0.0f);
    }
}

// ---------------- launcher ----------------
extern "C" void kernel_launch(void* const* d_in, const int* in_sizes, int n_in,
                              void* d_out, int out_size, void* d_ws, size_t ws_size,
                              hipStream_t stream) {
    (void)in_sizes; (void)n_in; (void)out_size; (void)ws_size;

    const float* x   = (const float*)d_in[0];   // [B*P, F]
    const float* pos = (const float*)d_in[1];   // [B*P, 3]
    // d_in[2] = batch (int64) — recomputed, not needed
    const float* W0 = (const float*)d_in[3];    // [67,128]
    const float* b0 = (const float*)d_in[4];
    const float* W1 = (const float*)d_in[5];    // [128,256]
    const float* b1 = (const float*)d_in[6];
    const float* W2 = (const float*)d_in[7];    // [256,256]
    const float* b2 = (const float*)d_in[8];

    // output layout: h [8192*256] | new_pos [8192*3] | new_batch [8192]
    float* out_h     = (float*)d_out;
    float* out_pos   = out_h + (size_t)BS * C2;
    float* out_batch = out_pos + (size_t)BS * 3;

    // workspace layout
    char* ws = (char*)d_ws;
    int*   idx     = (int*)(ws);                                  // 8192 * 4
    float* centers = (float*)(ws + 32768);                        // 8192*3*4 = 96 KB
    float* h0      = (float*)(ws + 131072);                       // 8192*68*4
    float* h1      = (float*)(ws + 2359296);                      // 8192*128*4
    float* h2      = (float*)(ws + 6553600);                      // 8192*256*4
    float* W0p     = (float*)(ws + 14942208);                     // 68*128*4

    // 1) FPS: one 1024-thread block per cloud (32 waves)
    fps_kernel<<<BB, 1024, 0, stream>>>(pos, idx);

    // 2) centers + new_pos + new_batch
    centers_kernel<<<(BS + 255) / 256, 256, 0, stream>>>(pos, idx, centers, out_pos, out_batch);

    // 3) pad W0 (independent of 1/2; stream-ordered is fine)
    pad_w0_kernel<<<(LDH * C0 + 127) / 128, 128, 0, stream>>>(W0, W0p);

    // 4) ball query + max pool: one wave per center, 4 waves/block
    ball_max_kernel<<<BS / 4, 128, 0, stream>>>(x, pos, centers, h0);

    // 5) MLP: three WMMA GEMMs (wave per 16x16 tile, 4 waves/block)
    // G0: [8192 x 68] x [68 x 128] -> h1
    gemm_bias_relu_wmma<<<(BS / 16) * (C0 / 16) / 4, 128, 0, stream>>>(h0, LDH, W0p, b0, h1, C0, LDH);
    // G1: [8192 x 128] x [128 x 256] -> h2
    gemm_bias_relu_wmma<<<(BS / 16) * (C1 / 16) / 4, 128, 0, stream>>>(h1, C0, W1, b1, h2, C1, C0);
    // G2: [8192 x 256] x [256 x 256] -> out_h
    gemm_bias_relu_wmma<<<(BS / 16) * (C2 / 16) / 4, 128, 0, stream>>>(h2, C1, W2, b2, out_h, C2, C1);
}